// _GradientBoostingBase_86517821215109
// MI455X (gfx1250) — compile-verified
//
#include <hip/hip_runtime.h>
#include <hip/hip_bf16.h>
#include <stdint.h>

// ---------------------------------------------------------------------------
// Gradient-boosted-tree inference for MI455X (gfx1250).
//   N=50000 rows, F=32 features, S=100 stages, T=3 classes, depth 6.
// v4: prep kernel packs {feat,thr} into 8B node records in d_ws; main kernel
// TDM-DMAs packed nodes (151KB) + leaves (77KB) into LDS, stages x rows in
// LDS, and walks all 300 trees per thread with one ds_load_b64 per node.
// Stage loop unrolled x4 -> 12 independent traversal chains per thread to
// cover LDS latency at the forced 1-block/WGP occupancy.
// ---------------------------------------------------------------------------

#define NROWS_PER_BLOCK 256
#define NFEAT     32
#define NSTAGES   100
#define NCLASS    3
#define NTREES    (NSTAGES * NCLASS)      // 300
#define NODES     63
#define LEAVES    64
#define NNODE_TOT (NTREES * NODES)        // 18900 nodes
#define NODE_DW   (NNODE_TOT * 2)         // 37800 dwords (8B per node)
#define LEAF_DW   (NTREES * LEAVES)       // 19200 dwords
#define LEAF_OFF_B  (NODE_DW * 4)                 // 151200
#define X_OFF_B     (LEAF_OFF_B + LEAF_DW * 4)    // 228000
#define XPITCH    33                              // bank-conflict scramble
#define SMEM_BYTES (X_OFF_B + NROWS_PER_BLOCK * XPITCH * 4)  // 261792 < 320KB

typedef unsigned int u32x4 __attribute__((ext_vector_type(4)));
typedef int          i32x8 __attribute__((ext_vector_type(8)));
typedef int          i32x4 __attribute__((ext_vector_type(4)));

#if defined(__gfx1250__)
// One TDM descriptor: 1-D copy of `ndw` dwords (ndw < 65536) global -> LDS.
// D# packing per CDNA5 ISA §8. This toolchain: 6-arg builtin form.
__device__ __forceinline__ void tdm_copy_dwords(const void* gptr,
                                                uint32_t lds_off,
                                                uint32_t ndw) {
  uint64_t ga = (uint64_t)(uintptr_t)gptr;
  u32x4 g0;
  g0[0] = 1u;                                           // count=1 (valid), user mode
  g0[1] = lds_off;                                      // lds_addr (bytes)
  g0[2] = (uint32_t)(ga & 0xFFFFFFFFu);                 // global_addr[31:0]
  g0[3] = (uint32_t)((ga >> 32) & 0x01FFFFFFu)          // global_addr[56:32]
        | 0x80000000u;                                  // type=2 ("image")
  i32x8 g1;
  g1[0] = (int)(0x2u << 16);                            // data_size=2 -> 4B; mask=0
  g1[1] = (int)((ndw & 0xFFFFu) << 16);                 // tensor_dim0[15:0]
  g1[2] = (int)(((ndw >> 16) & 0xFFFFu) | (1u << 16));  // tensor_dim0[31:16], tensor_dim1=1
  g1[3] = (int)((ndw & 0xFFFFu) << 16);                 // tensor_dim1[31:16]=0, tile_dim0=ndw
  g1[4] = 1;                                            // tile_dim1=1, tile_dim2=0
  g1[5] = (int)ndw;                                     // tensor_dim0_stride[31:0]
  g1[6] = 0;
  g1[7] = 0;
  i32x4 z4 = {0, 0, 0, 0};
  i32x8 z8 = {0, 0, 0, 0, 0, 0, 0, 0};
  __builtin_amdgcn_tensor_load_to_lds(g0, g1, z4, z4, z8, 0);
}
#endif

// Prep: interleave feat/thr into 8-byte node records in workspace.
__global__ __launch_bounds__(256)
void pack_nodes_kernel(const int* __restrict__ features,
                       const float* __restrict__ thresholds,
                       uint2* __restrict__ nodes, int n) {
  int i = blockIdx.x * 256 + threadIdx.x;
  if (i < n) {
    uint2 nd;
    nd.x = (unsigned)features[i];
    nd.y = __float_as_uint(thresholds[i]);
    nodes[i] = nd;                      // single global_store_b64, coalesced
  }
}

__global__ __launch_bounds__(NROWS_PER_BLOCK)
void gbt_eval_kernel(const float* __restrict__ x,
                     const int*   __restrict__ features,
                     const float* __restrict__ thresholds,
                     const uint2* __restrict__ ws_nodes,
                     const float* __restrict__ leaf_values,
                     const float* __restrict__ init_pred,
                     float* __restrict__ out, int N, int use_ws) {
  extern __shared__ char smem[];
  uint2* lds_nodes = (uint2*)(smem);                 // 18900 x 8B
  float* lds_leaf  = (float*)(smem + LEAF_OFF_B);    // 19200 x 4B
  float* lds_x     = (float*)(smem + X_OFF_B);       // 256 x 33 x 4B

  const int tid = threadIdx.x;
  const int row = blockIdx.x * NROWS_PER_BLOCK + tid;

#if defined(__gfx1250__)
  uint32_t base = (uint32_t)(uintptr_t)smem;         // low 32b == LDS byte offset
  if (use_ws) {
    if (tid < 32) {                                   // wave 0 issues the DMAs
      tdm_copy_dwords(ws_nodes,    base,              NODE_DW);
      tdm_copy_dwords(leaf_values, base + LEAF_OFF_B, LEAF_DW);
    }
  } else {
    if (tid < 32) tdm_copy_dwords(leaf_values, base + LEAF_OFF_B, LEAF_DW);
    for (int i = tid; i < NNODE_TOT; i += NROWS_PER_BLOCK) {
      uint2 nd;
      nd.x = (unsigned)features[i];
      nd.y = __float_as_uint(thresholds[i]);
      lds_nodes[i] = nd;
    }
  }
#else
  // Fallback (host parse / non-gfx1250): cooperative copy.
  for (int i = tid; i < NNODE_TOT; i += NROWS_PER_BLOCK) {
    uint2 nd;
    nd.x = (unsigned)features[i];
    nd.y = __float_as_uint(thresholds[i]);
    lds_nodes[i] = nd;
  }
  for (int i = tid; i < LEAF_DW; i += NROWS_PER_BLOCK)
    lds_leaf[i] = leaf_values[i];
#endif

  // Stage this thread's x-row into LDS (overlaps with the TDM transfer).
  if (row < N) {
    const float4* xr = (const float4*)(x + (size_t)row * NFEAT);
#pragma unroll
    for (int i = 0; i < NFEAT / 4; ++i) {
      float4 v = xr[i];
      lds_x[tid * XPITCH + 4 * i + 0] = v.x;
      lds_x[tid * XPITCH + 4 * i + 1] = v.y;
      lds_x[tid * XPITCH + 4 * i + 2] = v.z;
      lds_x[tid * XPITCH + 4 * i + 3] = v.w;
    }
  }

#if defined(__gfx1250__)
  __builtin_amdgcn_s_wait_tensorcnt(0);   // waves without TDM: trivially true
#endif
  __syncthreads();

  const float* myx = lds_x + tid * XPITCH;
  float acc0 = 0.f, acc1 = 0.f, acc2 = 0.f;

  // Unroll stages x4 -> 12 independent traversal chains in flight per thread
  // (occupancy is fixed at 2 waves/SIMD by the 262KB LDS footprint, so
  //  latency hiding must come from per-wave ILP).
#pragma unroll 4
  for (int s = 0; s < NSTAGES; ++s) {
#pragma unroll
    for (int t = 0; t < NCLASS; ++t) {
      const int m  = s * NCLASS + t;
      const int nb = m * NODES;
      int idx = 0;
#pragma unroll
      for (int d = 0; d < 6; ++d) {
        uint2 nd = lds_nodes[nb + idx];          // one ds_load_b64 per node
        float th = __uint_as_float(nd.y);
        float xv = myx[nd.x];
        idx = 2 * idx + ((xv <= th) ? 1 : 2);    // sklearn: left iff x <= thr
      }
      float lv = lds_leaf[m * LEAVES + (idx - NODES)];
      if (t == 0)      acc0 += lv;
      else if (t == 1) acc1 += lv;
      else             acc2 += lv;
    }
  }

  if (row < N) {
    out[row * 3 + 0] = init_pred[0] + 0.1f * acc0;
    out[row * 3 + 1] = init_pred[1] + 0.1f * acc1;
    out[row * 3 + 2] = init_pred[2] + 0.1f * acc2;
  }
}

extern "C" void kernel_launch(void* const* d_in, const int* in_sizes, int n_in,
                              void* d_out, int out_size, void* d_ws, size_t ws_size,
                              hipStream_t stream) {
  const float* x          = (const float*)d_in[0];   // [N,32] f32
  const int*   features   = (const int*)d_in[1];     // [100,3,63] i32
  const float* thresholds = (const float*)d_in[2];   // [100,3,63] f32
  const float* leaf_vals  = (const float*)d_in[3];   // [100,3,64] f32
  const float* init_pred  = (const float*)d_in[4];   // [3] f32
  float* out = (float*)d_out;                        // [N,3] f32

  const int N = in_sizes[0] / NFEAT;
  const int grid = (N + NROWS_PER_BLOCK - 1) / NROWS_PER_BLOCK;

  const size_t need = (size_t)NNODE_TOT * sizeof(uint2);   // 151200 B
  const int use_ws = (d_ws != nullptr && ws_size >= need) ? 1 : 0;

  static bool attr_set = false;
  if (!attr_set) {
    (void)hipFuncSetAttribute((const void*)gbt_eval_kernel,
                              hipFuncAttributeMaxDynamicSharedMemorySize,
                              SMEM_BYTES);
    attr_set = true;
  }

  if (use_ws) {
    pack_nodes_kernel<<<(NNODE_TOT + 255) / 256, 256, 0, stream>>>(
        features, thresholds, (uint2*)d_ws, NNODE_TOT);
  }

  gbt_eval_kernel<<<grid, NROWS_PER_BLOCK, SMEM_BYTES, stream>>>(
      x, features, thresholds, (const uint2*)d_ws, leaf_vals, init_pred,
      out, N, use_ws);
}